// MaximizeMutualInformationFamilyLoss_56693568307779
// MI455X (gfx1250) — compile-verified
//
#include <hip/hip_runtime.h>

typedef float v2f __attribute__((ext_vector_type(2)));
typedef float v8f __attribute__((ext_vector_type(8)));

#define NFAM 32
#define MREC 64

// ---------------------------------------------------------------------------
// Kernel 0: zero the workspace accumulators (S[32][64] + counts[32])
// ---------------------------------------------------------------------------
__global__ void __launch_bounds__(256) mi_zero_kernel(float* ws, int n) {
    for (int i = blockIdx.x * blockDim.x + threadIdx.x; i < n;
         i += gridDim.x * blockDim.x)
        ws[i] = 0.0f;
}

// ---------------------------------------------------------------------------
// One 4-row group: load A one-hot + B fragments, 8 f32 WMMAs.
// ---------------------------------------------------------------------------
__device__ __forceinline__ void mi_group(const float* __restrict__ act,
                                         const int* __restrict__ fids,
                                         long g, int l16, int k0,
                                         v8f acc[2][4]) {
    const long base = g << 2;

    // Family ids for this lane's two K slots (k0, k0+1): one b64 load.
    const int2 fid = *(const int2*)(fids + base + k0);

    // B fragments: B(K, Ncol), K = k0 + vgpr, Ncol = mt*16 + l16.
    const float* rowp = act + (base + k0) * MREC + l16;
    float b0[4], b1[4];
#pragma unroll
    for (int mt = 0; mt < 4; ++mt) {
        b0[mt] = rowp[mt * 16];         // row base+k0
        b1[mt] = rowp[mt * 16 + MREC];  // row base+k0+1
    }

#pragma unroll
    for (int ft = 0; ft < 2; ++ft) {
        // A fragment: one-hot indicator, A(M=l16, K=k0/k0+1). Exact in f32.
        const int fam = ft * 16 + l16;
        v2f a;
        a.x = (fid.x == fam) ? 1.0f : 0.0f;
        a.y = (fid.y == fam) ? 1.0f : 0.0f;
#pragma unroll
        for (int mt = 0; mt < 4; ++mt) {
            v2f b;
            b.x = b0[mt];
            b.y = b1[mt];
            acc[ft][mt] = __builtin_amdgcn_wmma_f32_16x16x4_f32(
                /*neg_a=*/false, a, /*neg_b=*/false, b,
                /*c_mod=*/(short)0, acc[ft][mt],
                /*reuse_a=*/false, /*reuse_b=*/false);
        }
    }
}

// ---------------------------------------------------------------------------
// Kernel 1: segmented sum via one-hot WMMA  (S = onehot(F)^T * A), plus
// family histogram. f32 WMMA (16x16x4) keeps full f32 precision.
// ---------------------------------------------------------------------------
__global__ void __launch_bounds__(256) mi_accum_kernel(
    const float* __restrict__ act,   // [N][64] row-major
    const int*   __restrict__ fids,  // [N]
    float* __restrict__ gS,          // [32*64]
    float* __restrict__ gCnt,        // [32]
    int N) {
    __shared__ float ldsS[NFAM * MREC];
    __shared__ int   ldsH[NFAM];

    for (int i = threadIdx.x; i < NFAM * MREC; i += blockDim.x) ldsS[i] = 0.0f;
    if (threadIdx.x < NFAM) ldsH[threadIdx.x] = 0;
    __syncthreads();

    const int lane          = threadIdx.x & 31;
    const int waves_per_blk = blockDim.x >> 5;
    // Force the wave index into an SGPR: the g-loop becomes a scalar loop,
    // guaranteeing EXEC is all-1s at every WMMA (ISA requirement) and
    // removing exec-mask bookkeeping from the hot loop.
    const int wave =
        __builtin_amdgcn_readfirstlane(blockIdx.x * waves_per_blk +
                                       (threadIdx.x >> 5));
    const int nwaves = gridDim.x * waves_per_blk;

    const int half = lane >> 4;   // 0: lanes 0-15, 1: lanes 16-31
    const int l16  = lane & 15;
    const int k0   = half << 1;   // K held in VGPR0 of A/B fragments (0 or 2)

    // 8 accumulator tiles: [f-tile 0..1][m-tile 0..3], each 16x16 f32.
    v8f acc[2][4];
#pragma unroll
    for (int ft = 0; ft < 2; ++ft)
#pragma unroll
        for (int mt = 0; mt < 4; ++mt) acc[ft][mt] = 0.0f;

    const int ngroups = N >> 2;          // K=4 rows per group
    const int step    = nwaves << 1;     // 2 groups (2 KB) per iteration
    int g = wave << 1;
    for (; g + 1 < ngroups; g += step) { // unroll x2: 18 loads then 16 WMMAs
        mi_group(act, fids, (long)g,     l16, k0, acc);
        mi_group(act, fids, (long)g + 1, l16, k0, acc);
    }
    if (g < ngroups)                     // uniform tail (odd group count)
        mi_group(act, fids, (long)g, l16, k0, acc);

    // Flush D tiles into the block-level LDS accumulator.
    // D layout: VGPR v, this lane -> row M = v + 8*half, col = l16.
#pragma unroll
    for (int ft = 0; ft < 2; ++ft)
#pragma unroll
        for (int mt = 0; mt < 4; ++mt)
#pragma unroll
            for (int v = 0; v < 8; ++v) {
                const int f = ft * 16 + v + 8 * half;
                const int m = mt * 16 + l16;
                atomicAdd(&ldsS[f * MREC + m], acc[ft][mt][v]);
            }

    // Family histogram (2 MB stream, negligible vs 128 MB).
    for (int i = blockIdx.x * blockDim.x + threadIdx.x; i < N;
         i += gridDim.x * blockDim.x)
        atomicAdd(&ldsH[fids[i]], 1);

    __syncthreads();

    // Block -> global: 2048 + 32 atomic f32 adds per block.
    for (int i = threadIdx.x; i < NFAM * MREC; i += blockDim.x)
        unsafeAtomicAdd(&gS[i], ldsS[i]);
    if (threadIdx.x < NFAM)
        unsafeAtomicAdd(&gCnt[threadIdx.x], (float)ldsH[threadIdx.x]);
}

// ---------------------------------------------------------------------------
// Kernel 2: entropy epilogue (tiny, one block of 64 threads).
// ---------------------------------------------------------------------------
__global__ void __launch_bounds__(64) mi_finalize_kernel(
    const float* __restrict__ gS, const float* __restrict__ gCnt,
    float* __restrict__ out, float Nf) {
    __shared__ float sh_hf[NFAM];
    __shared__ float sh_ha[MREC];
    const int t = threadIdx.x;

    if (t < NFAM) {
        const float cnt = gCnt[t];
        float hf = 0.0f;
        if (cnt > 0.0f) {
            for (int m = 0; m < MREC; ++m) {
                const float p1 = gS[t * MREC + m] / cnt;
                const float p0 = 1.0f - p1;
                hf -= logf(p0 * p0 + p1 * p1);
            }
        }
        sh_hf[t] = hf;
    }
    {
        float colsum = 0.0f;
        for (int f = 0; f < NFAM; ++f) colsum += gS[f * MREC + t];
        const float p1 = colsum / Nf;
        const float p0 = 1.0f - p1;
        sh_ha[t] = -logf(p0 * p0 + p1 * p1);
    }
    __syncthreads();

    if (t == 0) {
        float ha = 0.0f;
        for (int m = 0; m < MREC; ++m) ha += sh_ha[m];
        float s = 0.0f, nfam = 0.0f;
        for (int f = 0; f < NFAM; ++f)
            if (gCnt[f] > 0.0f) { s += sh_hf[f]; nfam += 1.0f; }
        out[0] = s / nfam - ha;
    }
}

// ---------------------------------------------------------------------------
extern "C" void kernel_launch(void* const* d_in, const int* in_sizes, int n_in,
                              void* d_out, int out_size, void* d_ws,
                              size_t ws_size, hipStream_t stream) {
    const float* act  = (const float*)d_in[0];
    const int*   fids = (const int*)d_in[1];
    const int N = in_sizes[1];  // family_ids count = number of samples

    float* gS   = (float*)d_ws;        // [32*64]
    float* gCnt = gS + NFAM * MREC;    // [32]

    mi_zero_kernel<<<1, 256, 0, stream>>>(gS, NFAM * MREC + NFAM);
    mi_accum_kernel<<<512, 256, 0, stream>>>(act, fids, gS, gCnt, N);
    mi_finalize_kernel<<<1, 64, 0, stream>>>(gS, gCnt, (float*)d_out, (float)N);
}